// NormalizingFlow_29265907155419
// MI455X (gfx1250) — compile-verified
//
#include <hip/hip_runtime.h>
#include <math.h>

// ---------------- Problem constants (from reference) ----------------
#define BN      262144
#define DN      64
#define LN      4
#define KN      8
#define BOUNDF  3.0f
#define MIN_BWF 1e-3f
#define MIN_BHF 1e-3f
#define MIN_DF  1e-3f
#define MIN_LAM 0.025f
#define EPSF    1e-6f

// ---------------- Table layout ----------------
// Per layer:
//   search knots : [k=0..6][d=0..63]            -> 448 floats (cumw[k+1]+EPS)
//   coeff table  : [f=0..12][bin=0..7][d=0..63] -> 6656 floats
#define NFIELD   13
#define SEARCH_F (7 * DN)                 // 448
#define TABLE_F  (NFIELD * KN * DN)       // 6656
#define LAYER_F  (SEARCH_F + TABLE_F)     // 7104
#define ALL_F    (LN * LAYER_F)           // 28416 floats = 113664 B
#define SMEM_FLOATS (ALL_F + 16)          // + 16 parity-buffered reduction partials

#define NBLK   2048
#define TPB    256
#define STRIDE (NBLK * TPB)               // 524288 (multiple of 64 -> fixed d per thread)
#define NITER  ((BN * DN) / STRIDE)       // 32

// ---------------- TDM availability ----------------
#if defined(__AMDGCN__) && __has_builtin(__builtin_amdgcn_tensor_load_to_lds) && \
    __has_builtin(__builtin_amdgcn_s_wait_tensorcnt)
#define HAVE_TDM 1
#else
#define HAVE_TDM 0
#endif

typedef unsigned int u32x4 __attribute__((ext_vector_type(4)));
typedef int          i32x8 __attribute__((ext_vector_type(8)));
typedef int          i32x4 __attribute__((ext_vector_type(4)));

// ---------------- Kernel 2: streaming 4-layer flow evaluation ----------------
// (placed first so the disasm snippet shows the TDM sequence)
__global__ void __launch_bounds__(TPB)
nf_flow(const float* __restrict__ x, const float* __restrict__ tb,
        float* __restrict__ outx, float* __restrict__ outld) {
    extern __shared__ float smem[];

#if HAVE_TDM
    // ---- Tensor Data Mover: DMA the whole 113664-byte table global->LDS ----
    // 1-D copy of 14208 8-byte units. D# layout per CDNA5 ISA ch.7/8.
    if (threadIdx.x < 32) {   // one wave issues the TDM op (EXEC ignored by TDM)
        const unsigned U8 = (unsigned)(ALL_F / 2);           // 14208 8-byte elements
        const unsigned long long ga = (unsigned long long)(uintptr_t)tb;
        const unsigned lds_base =
            (unsigned)(size_t)(__attribute__((address_space(3))) float*)smem;

        u32x4 g0;
        g0.x = 1u;                                            // count=1 (valid, user desc)
        g0.y = lds_base;                                      // lds_addr (bytes)
        g0.z = (unsigned)ga;                                  // global_addr[31:0]
        g0.w = ((unsigned)(ga >> 32) & 0x01FFFFFFu)           // global_addr[56:32]
               | 0x80000000u;                                 // type=2 ("image")

        i32x8 g1;
        g1[0] = (3 << 16);                                    // data_size=3 (8 bytes)
        g1[1] = (int)((U8 & 0xFFFFu) << 16);                  // tensor_dim0[15:0]
        g1[2] = (int)((U8 >> 16) | (1u << 16));               // tensor_dim0[31:16], tensor_dim1=1
        g1[3] = (int)((U8 & 0xFFFFu) << 16);                  // tile_dim0 = U8 (tensor_dim1 hi = 0)
        g1[4] = 0;                                            // tile_dim1=0 (unused), tile_dim2=0
        g1[5] = (int)U8;                                      // tensor_dim0_stride[31:0]
        g1[6] = 0;                                            // stride0[47:32], stride1 lo
        g1[7] = 0;

        i32x4 gz4 = {0, 0, 0, 0};                             // groups 2/3 unused (<=2D)
        i32x8 gz8 = {0, 0, 0, 0, 0, 0, 0, 0};
        // 6-arg form: (u32x4 g0, i32x8 g1, i32x4, i32x4, i32x8, i32 cpol)
        __builtin_amdgcn_tensor_load_to_lds(g0, g1, gz4, gz4, gz8, 0);
        __builtin_amdgcn_s_wait_tensorcnt(0);                 // s_wait_tensorcnt 0
    }
    __syncthreads();
#else
    {   // fallback: b128 copy loop
        const float4* __restrict__ g4 = (const float4*)tb;
        float4* s4 = (float4*)smem;
        for (int i = threadIdx.x; i < ALL_F / 4; i += TPB) s4[i] = g4[i];
    }
    __syncthreads();
#endif

    const int d = threadIdx.x & (DN - 1);     // fixed dimension per thread
    float kn[LN][7];                          // register-resident bin-search knots
#pragma unroll
    for (int l = 0; l < LN; ++l)
#pragma unroll
        for (int k = 0; k < 7; ++k) kn[l][k] = smem[l * LAYER_F + k * DN + d];

    const int tid = blockIdx.x * TPB + threadIdx.x;
    const int wv = threadIdx.x >> 5;          // wave id in block (0..7)
    float* partial = smem + ALL_F;            // 16 floats: two parity sets of 8

    for (int it = 0; it < NITER; ++it) {
        const int idx = tid + it * STRIDE;
        if (it + 1 < NITER) __builtin_prefetch(x + idx + STRIDE, 0, 0);  // global_prefetch_b8

        float xv = __builtin_nontemporal_load(x + idx);      // streamed once, keep L2 clean
        float lad = 0.0f;

#pragma unroll
        for (int l = 0; l < LN; ++l) {
            const float* tbl = smem + l * LAYER_F + SEARCH_F;
            const bool inside = (xv >= -BOUNDF) && (xv <= BOUNDF);
            const float xc = fminf(fmaxf(xv, -BOUNDF), BOUNDF);
            int bin = 0;
#pragma unroll
            for (int k = 0; k < 7; ++k) bin += (xc >= kn[l][k]) ? 1 : 0;

            const float* e = tbl + bin * DN + d;   // lanes -> consecutive LDS banks
            const float icw   = e[0 * 512];
            const float inviw = e[1 * 512];
            const float il    = e[2 * 512];
            const float th    = (xc - icw) * inviw;
            const bool  lo    = (th <= il);
            const float nA = lo ? e[3 * 512]  : e[7 * 512];
            const float nB = lo ? e[4 * 512]  : e[8 * 512];
            const float dA = lo ? e[5 * 512]  : e[9 * 512];
            const float dB = lo ? e[6 * 512]  : e[10 * 512];
            const float lg = lo ? e[11 * 512] : e[12 * 512];

            const float num = fmaf(nB, th, nA);
            const float den = fmaf(dB, th, dA);
            const float ov  = num / den;
            const float la  = lg - 2.0f * logf(fabsf(den));
            xv  = inside ? ov : xv;
            lad = inside ? (lad + la) : lad;
        }

        __builtin_nontemporal_store(xv, outx + idx);

        // log-det: deterministic reduction over D=64 (2 wave32 partials + pair add).
        // Parity-alternating partial slots -> single barrier per iteration.
        float r = lad;
#pragma unroll
        for (int off = 16; off > 0; off >>= 1) r += __shfl_xor(r, off, 32);
        const int pbase = (it & 1) << 3;
        if ((threadIdx.x & 31) == 0) partial[pbase + wv] = r;
        __syncthreads();
        if ((threadIdx.x & 63) == 0)
            __builtin_nontemporal_store(partial[pbase + wv] + partial[pbase + wv + 1],
                                        outld + (idx >> 6));
    }
}

// ---------------- Small math helpers ----------------
__device__ __forceinline__ float softplus_f(float v) {
    return fmaxf(v, 0.0f) + log1pf(expf(-fabsf(v)));   // jax.nn.softplus
}
__device__ __forceinline__ float sigmoid_f(float v) {
    return 1.0f / (1.0f + expf(-v));
}

__device__ __forceinline__ void norm_bins(const float* __restrict__ p, float minb,
                                          float* __restrict__ o) {
    float m = p[0];
#pragma unroll
    for (int k = 1; k < KN; ++k) m = fmaxf(m, p[k]);
    float e[KN];
    float s = 0.0f;
#pragma unroll
    for (int k = 0; k < KN; ++k) { e[k] = expf(p[k] - m); s += e[k]; }
    float inv = 1.0f / s;
#pragma unroll
    for (int k = 0; k < KN; ++k) o[k] = minb + (1.0f - minb * (float)KN) * (e[k] * inv);
}

__device__ __forceinline__ void make_knots(const float* __restrict__ len,
                                           float* __restrict__ cum,
                                           float* __restrict__ width) {
    cum[0] = -BOUNDF;
    float c = 0.0f;
#pragma unroll
    for (int k = 0; k < KN; ++k) { c += len[k]; cum[k + 1] = 2.0f * BOUNDF * c - BOUNDF; }
    cum[KN] = BOUNDF;
#pragma unroll
    for (int k = 0; k < KN; ++k) width[k] = cum[k + 1] - cum[k];
}

// ---------------- Kernel 1: per-(layer,dim) spline coefficient tables ----------------
__global__ void nf_preprocess(const float* __restrict__ uw, const float* __restrict__ uh,
                              const float* __restrict__ ud, const float* __restrict__ ul,
                              float* __restrict__ tb) {
    int t = blockIdx.x * blockDim.x + threadIdx.x;
    if (t >= LN * DN) return;
    int l = t / DN, d = t % DN;

    const float* pw = uw + (size_t)(l * DN + d) * KN;
    const float* ph = uh + (size_t)(l * DN + d) * KN;
    const float* pd = ud + (size_t)(l * DN + d) * (KN - 1);
    const float* pl = ul + (size_t)(l * DN + d) * KN;

    float w[KN], h[KN], cw[KN + 1], ch[KN + 1], wid[KN], hei[KN], dv[KN + 1], lm[KN];
    norm_bins(pw, MIN_BWF, w);
    norm_bins(ph, MIN_BHF, h);
    make_knots(w, cw, wid);
    make_knots(h, ch, hei);
    dv[0] = 1.0f; dv[KN] = 1.0f;
#pragma unroll
    for (int k = 0; k < KN - 1; ++k) dv[k + 1] = MIN_DF + softplus_f(pd[k]);
#pragma unroll
    for (int k = 0; k < KN; ++k) lm[k] = (1.0f - 2.0f * MIN_LAM) * sigmoid_f(pl[k]) + MIN_LAM;

    float* base = tb + (size_t)l * LAYER_F;
#pragma unroll
    for (int k = 0; k < 7; ++k) base[k * DN + d] = cw[k + 1] + EPSF;

    float* T = base + SEARCH_F;
#pragma unroll
    for (int b = 0; b < KN; ++b) {
        float iw = wid[b], inviw = 1.0f / iw;
        float icw = cw[b], ich = ch[b], ih = hei[b];
        float idelta = ih * inviw;
        float id0 = dv[b], id1 = dv[b + 1], il = lm[b];
        float wb = sqrtf(id0 / id1);                         // wa = 1
        float wc = (il * id0 + (1.0f - il) * wb * id1) / idelta;
        float ya = ich, yb = ich + ih;
        float yc = ((1.0f - il) * ya + il * wb * yb) / ((1.0f - il) + il * wb);

        float* e = T + b * DN + d;                           // field stride = KN*DN = 512
        e[0 * 512]  = icw;
        e[1 * 512]  = inviw;
        e[2 * 512]  = il;
        e[3 * 512]  = ya * il;                 // nA_lo
        e[4 * 512]  = wc * yc - ya;            // nB_lo
        e[5 * 512]  = il;                      // dA_lo
        e[6 * 512]  = wc - 1.0f;               // dB_lo
        e[7 * 512]  = wc * yc - wb * yb * il;  // nA_hi
        e[8 * 512]  = wb * yb - wc * yc;       // nB_hi
        e[9 * 512]  = wc - wb * il;            // dA_hi
        e[10 * 512] = wb - wc;                 // dB_hi
        e[11 * 512] = logf(wc * il * (yc - ya) * inviw);               // log dnum, lower
        e[12 * 512] = logf(wb * wc * (1.0f - il) * (yb - yc) * inviw); // log dnum, upper
    }
}

// ---------------- Host entry ----------------
extern "C" void kernel_launch(void* const* d_in, const int* in_sizes, int n_in,
                              void* d_out, int out_size, void* d_ws, size_t ws_size,
                              hipStream_t stream) {
    (void)in_sizes; (void)n_in; (void)out_size; (void)ws_size;
    const float* x  = (const float*)d_in[0];
    const float* uw = (const float*)d_in[1];
    const float* uh = (const float*)d_in[2];
    const float* ud = (const float*)d_in[3];
    const float* ul = (const float*)d_in[4];
    float* tb    = (float*)d_ws;                       // 28416 floats (111 KB) scratch
    float* outx  = (float*)d_out;                      // (B, D)
    float* outld = outx + (size_t)BN * DN;             // (B,)

    nf_preprocess<<<1, 256, 0, stream>>>(uw, uh, ud, ul, tb);
    nf_flow<<<NBLK, TPB, SMEM_FLOATS * sizeof(float), stream>>>(x, tb, outx, outld);
}